// HierarchicalGraphModule_64415919505529
// MI455X (gfx1250) — compile-verified
//
#include <hip/hip_runtime.h>
#include <cstdint>
#include <cstddef>

// ---------------------------------------------------------------------------
// Problem constants (match reference)
// ---------------------------------------------------------------------------
#define N_NODES 50000
#define N_EDGES 800000
#define DH      128
#define DOUT    64
#define EPS_BN  1e-5f

typedef __attribute__((ext_vector_type(2))) float v2f;
typedef __attribute__((ext_vector_type(8))) float v8f;

// ---------------------------------------------------------------------------
// Utility kernels
// ---------------------------------------------------------------------------
__global__ void zero_i32_kernel(int* __restrict__ p, int n) {
    int i = blockIdx.x * blockDim.x + threadIdx.x;
    if (i < n) p[i] = 0;
}

__global__ void zero_f32_kernel(float* __restrict__ p, int n) {
    int i = blockIdx.x * blockDim.x + threadIdx.x;
    if (i < n) p[i] = 0.0f;
}

// ---------------------------------------------------------------------------
// Graph normalization + CSR construction (rebuilt deterministically per call)
// ---------------------------------------------------------------------------
__global__ void edge_hist_kernel(const int* __restrict__ src, const int* __restrict__ dst,
                                 int* __restrict__ cntSrc, int* __restrict__ cntDst, int ne) {
    int e = blockIdx.x * blockDim.x + threadIdx.x;
    if (e >= ne) return;
    int s = src[e], d = dst[e];
    if (s != d) {
        atomicAdd(&cntSrc[s], 1);
        atomicAdd(&cntDst[d], 1);
    }
}

__global__ void dis_kernel(const int* __restrict__ cntSrc, float* __restrict__ dis, int n) {
    int i = blockIdx.x * blockDim.x + threadIdx.x;
    if (i >= n) return;
    int c = cntSrc[i];
    dis[i] = (c > 0) ? rsqrtf((float)c) : 0.0f;
}

// Single-block exclusive scan (n ~ 50k -> fine)
__global__ void scan_kernel(const int* __restrict__ cnt, int* __restrict__ rowstart, int n) {
    __shared__ int buf[256];
    __shared__ int carry;
    int tid = threadIdx.x;
    if (tid == 0) carry = 0;
    __syncthreads();
    for (int base = 0; base < n; base += 256) {
        int x = (base + tid < n) ? cnt[base + tid] : 0;
        buf[tid] = x;
        __syncthreads();
        for (int off = 1; off < 256; off <<= 1) {
            int v = (tid >= off) ? buf[tid - off] : 0;
            __syncthreads();
            buf[tid] += v;
            __syncthreads();
        }
        if (base + tid < n) rowstart[base + tid] = carry + buf[tid] - x;
        __syncthreads();
        if (tid == 0) carry += buf[255];
        __syncthreads();
    }
    if (threadIdx.x == 0) rowstart[n] = carry;
}

__global__ void edge_scatter_kernel(const int* __restrict__ src, const int* __restrict__ dst,
                                    const float* __restrict__ dis,
                                    const int* __restrict__ rowstart, int* __restrict__ cursor,
                                    int* __restrict__ esrc, float* __restrict__ ew, int ne) {
    int e = blockIdx.x * blockDim.x + threadIdx.x;
    if (e >= ne) return;
    int s = src[e], d = dst[e];
    if (s == d) return;
    int pos = rowstart[d] + atomicAdd(&cursor[d], 1);
    esrc[pos] = s;
    ew[pos]   = -dis[s] * dis[d];
}

// ---------------------------------------------------------------------------
// Gather SpMM: out[row,:] = sum_e w_e * v[esrc_e,:]   (one wave32 per row,
// 4 features per lane, register accumulation, no atomics)
// ---------------------------------------------------------------------------
__global__ __launch_bounds__(256)
void spmm_csr_kernel(const int* __restrict__ rowstart, const int* __restrict__ esrc,
                     const float* __restrict__ ew, const float* __restrict__ v,
                     float* __restrict__ out, int n) {
    int row = blockIdx.x * 8 + (threadIdx.x >> 5);
    if (row >= n) return;
    int lane = threadIdx.x & 31;
    int beg = rowstart[row], end = rowstart[row + 1];
    float a0 = 0.f, a1 = 0.f, a2 = 0.f, a3 = 0.f;
    for (int i = beg; i < end; ++i) {
        int   s = esrc[i];
        float w = ew[i];
        if (i + 1 < end) {
            int s2 = esrc[i + 1];
            __builtin_prefetch(v + (size_t)s2 * DH + lane * 4, 0, 1);
        }
        const float4 xv = *reinterpret_cast<const float4*>(v + (size_t)s * DH + lane * 4);
        a0 = fmaf(w, xv.x, a0);
        a1 = fmaf(w, xv.y, a1);
        a2 = fmaf(w, xv.z, a2);
        a3 = fmaf(w, xv.w, a3);
    }
    float4 r; r.x = a0; r.y = a1; r.z = a2; r.w = a3;
    *reinterpret_cast<float4*>(out + (size_t)row * DH + lane * 4) = r;
}

// ---------------------------------------------------------------------------
// WMMA GEMM with M register-blocking:
//   out[(16*MTILES) x NCOLS tile] = sum_t A_t @ W_t(128xNCOLS) + bias
// Each wave owns a 16-col slice and MTILES row-tiles; one B fragment feeds
// MTILES independent V_WMMA_F32_16X16X4_F32 accumulator chains.
// NTERMS==3 fuses A2_eff = 2*A2 - A0 (Chebyshev T2) during LDS staging.
// ---------------------------------------------------------------------------
template <int NTERMS, int NCOLS, int MTILES>
__global__ __launch_bounds__(NCOLS * 2)
void cheb_gemm_kernel(const float* __restrict__ A0, const float* __restrict__ A1,
                      const float* __restrict__ A2, const float* __restrict__ W,
                      const float* __restrict__ bias, float* __restrict__ out, int nrows) {
    constexpr int LDSS = 132;                 // padded row stride (floats)
    constexpr int MR   = 16 * MTILES;         // rows per block
    __shared__ float As[MR * LDSS];

    const int tid     = threadIdx.x;
    const int wave    = tid >> 5;
    const int lane    = tid & 31;
    const int half    = lane >> 4;            // 0: lanes 0-15, 1: lanes 16-31
    const int laneN   = lane & 15;
    const int rowBase = blockIdx.x * MR;
    const int colBase = wave * 16;

    v8f c[MTILES];
#pragma unroll
    for (int m = 0; m < MTILES; ++m) c[m] = {};

    for (int tt = 0; tt < NTERMS; ++tt) {
        __syncthreads();                       // protect LDS reuse across terms
        const float* Ap = (tt == 0) ? A0 : (tt == 1) ? A1 : A2;
        // ---- stage MR x 128 A tile into LDS (coalesced, row-clamped) ----
        for (int e = tid; e < MR * 128; e += NCOLS * 2) {
            int r = e >> 7, col = e & 127;
            int gr = rowBase + r;
            if (gr > nrows - 1) gr = nrows - 1;     // clamp; stores are guarded
            size_t g = (size_t)gr * 128 + col;
            float val;
            if (NTERMS == 3 && tt == 2) val = 2.0f * A2[g] - A0[g];   // T2 = 2*L*T1 - T0
            else                        val = Ap[g];
            As[r * LDSS + col] = val;
        }
        __syncthreads();

        const float* __restrict__ Wt = W + (size_t)tt * 128 * NCOLS;
        // ---- K loop: 128 in steps of 4 ----
#pragma unroll 8
        for (int kk = 0; kk < 128; kk += 4) {
            // B frag: K pair (kk+2h, kk+2h+1) for column colBase+laneN
            v2f b;
            b.x = Wt[(size_t)(kk + 2 * half)     * NCOLS + colBase + laneN];
            b.y = Wt[(size_t)(kk + 2 * half + 1) * NCOLS + colBase + laneN];
#pragma unroll
            for (int m = 0; m < MTILES; ++m) {
                // A frag (ISA layout): lane half h holds K = kk+2h, kk+2h+1 of row laneN
                v2f a = *reinterpret_cast<const v2f*>(
                    &As[(m * 16 + laneN) * LDSS + kk + 2 * half]);
                c[m] = __builtin_amdgcn_wmma_f32_16x16x4_f32(false, a, false, b,
                                                             (short)0, c[m], false, false);
            }
        }
    }

    // ---- epilogue: bias + guarded store (C/D layout: VGPR i -> M = i + 8*half) ----
    float bb = bias[colBase + laneN];
#pragma unroll
    for (int m = 0; m < MTILES; ++m) {
#pragma unroll
        for (int i = 0; i < 8; ++i) {
            int row = rowBase + m * 16 + i + 8 * half;
            if (row < nrows)
                out[(size_t)row * NCOLS + colBase + laneN] = c[m][i] + bb;
        }
    }
}

// ---------------------------------------------------------------------------
// BatchNorm statistics (column sums over all rows), scale/shift finalize
// ---------------------------------------------------------------------------
__global__ __launch_bounds__(256)
void bn_stats_kernel(const float* __restrict__ y, float* __restrict__ sum,
                     float* __restrict__ sumsq, int n) {
    int col  = threadIdx.x & 127;
    int half = threadIdx.x >> 7;
    int row0 = blockIdx.x * 128;
    float s = 0.f, q = 0.f;
    int rend = min(row0 + 128, n);
    for (int r = row0 + half; r < rend; r += 2) {
        float v = y[(size_t)r * DH + col];
        s += v;
        q += v * v;
    }
    __shared__ float sh[256], shq[256];
    sh[threadIdx.x]  = s;
    shq[threadIdx.x] = q;
    __syncthreads();
    if (half == 0) {
        atomicAdd(&sum[col],   s + sh[threadIdx.x + 128]);
        atomicAdd(&sumsq[col], q + shq[threadIdx.x + 128]);
    }
}

__global__ void bn_finalize_kernel(const float* __restrict__ sum, const float* __restrict__ sumsq,
                                   const float* __restrict__ g, const float* __restrict__ b,
                                   float* __restrict__ scale, float* __restrict__ shift,
                                   float invN) {
    int c = threadIdx.x;            // 128 threads
    float m   = sum[c] * invN;
    float var = fmaxf(sumsq[c] * invN - m * m, 0.0f);
    float sc  = g[c] * rsqrtf(var + EPS_BN);
    scale[c] = sc;
    shift[c] = b[c] - m * sc;
}

// float4-vectorized BN scale/shift + ReLU, in place
__global__ void bn_relu_kernel(float* __restrict__ y, const float* __restrict__ scale,
                               const float* __restrict__ shift, int n4) {
    int i = blockIdx.x * blockDim.x + threadIdx.x;
    if (i >= n4) return;
    int c = (i * 4) & 127;
    float4 v  = *reinterpret_cast<float4*>(y + (size_t)i * 4);
    float4 sc = *reinterpret_cast<const float4*>(scale + c);
    float4 sf = *reinterpret_cast<const float4*>(shift + c);
    v.x = fmaxf(fmaf(v.x, sc.x, sf.x), 0.0f);
    v.y = fmaxf(fmaf(v.y, sc.y, sf.y), 0.0f);
    v.z = fmaxf(fmaf(v.z, sc.z, sf.z), 0.0f);
    v.w = fmaxf(fmaf(v.w, sc.w, sf.w), 0.0f);
    *reinterpret_cast<float4*>(y + (size_t)i * 4) = v;
}

// ---------------------------------------------------------------------------
// Fused: h = relu(LayerNorm(BN(z) + residual))   (one wave32 per row)
// ---------------------------------------------------------------------------
__global__ __launch_bounds__(256)
void bn_res_ln_relu_kernel(const float* __restrict__ z, const float* __restrict__ res,
                           const float* __restrict__ scale, const float* __restrict__ shift,
                           const float* __restrict__ lg, const float* __restrict__ lb,
                           float* __restrict__ out, int n) {
    int row = blockIdx.x * 8 + (threadIdx.x >> 5);
    if (row >= n) return;
    int lane = threadIdx.x & 31;
    size_t base = (size_t)row * DH + lane * 4;

    float4 v  = *reinterpret_cast<const float4*>(z + base);
    float4 r  = *reinterpret_cast<const float4*>(res + base);
    float4 sc = *reinterpret_cast<const float4*>(scale + lane * 4);
    float4 sf = *reinterpret_cast<const float4*>(shift + lane * 4);

    v.x = fmaf(v.x, sc.x, sf.x) + r.x;
    v.y = fmaf(v.y, sc.y, sf.y) + r.y;
    v.z = fmaf(v.z, sc.z, sf.z) + r.z;
    v.w = fmaf(v.w, sc.w, sf.w) + r.w;

    float s = v.x + v.y + v.z + v.w;
#pragma unroll
    for (int m = 16; m >= 1; m >>= 1) s += __shfl_xor(s, m, 32);
    float mean = s * (1.0f / 128.0f);

    float dx = v.x - mean, dy = v.y - mean, dz = v.z - mean, dw = v.w - mean;
    float q = dx * dx + dy * dy + dz * dz + dw * dw;
#pragma unroll
    for (int m = 16; m >= 1; m >>= 1) q += __shfl_xor(q, m, 32);
    float inv = rsqrtf(q * (1.0f / 128.0f) + EPS_BN);

    float4 g4 = *reinterpret_cast<const float4*>(lg + lane * 4);
    float4 b4 = *reinterpret_cast<const float4*>(lb + lane * 4);

    float4 o;
    o.x = fmaxf(fmaf(dx * inv, g4.x, b4.x), 0.0f);
    o.y = fmaxf(fmaf(dy * inv, g4.y, b4.y), 0.0f);
    o.z = fmaxf(fmaf(dz * inv, g4.z, b4.z), 0.0f);
    o.w = fmaxf(fmaf(dw * inv, g4.w, b4.w), 0.0f);
    *reinterpret_cast<float4*>(out + base) = o;
}

// ---------------------------------------------------------------------------
// Host-side orchestration
// ---------------------------------------------------------------------------
struct BlockParams {
    const float *b1, *b2, *bn1_b, *bn1_g, *bn2_b, *bn2_g, *ln_b, *ln_g, *w1, *w2;
};

extern "C" void kernel_launch(void* const* d_in, const int* in_sizes, int n_in,
                              void* d_out, int out_size, void* d_ws, size_t ws_size,
                              hipStream_t stream) {
    (void)in_sizes; (void)n_in; (void)out_size; (void)ws_size;

    const float* x   = (const float*)d_in[0];
    const int*   src = (const int*)d_in[1];
    const int*   dst = src + N_EDGES;
    // d_in[2] = edge_weight: unused by the reference module.

    // params flattened in jax pytree (sorted-key) order:
    // blocks[0..2]{b1,b2,bn1_b,bn1_g,bn2_b,bn2_g,ln_b,ln_g,w1,w2}, fc_b, fc_w
    BlockParams B[3];
    int p = 3;
    for (int i = 0; i < 3; ++i) {
        B[i].b1    = (const float*)d_in[p++];
        B[i].b2    = (const float*)d_in[p++];
        B[i].bn1_b = (const float*)d_in[p++];
        B[i].bn1_g = (const float*)d_in[p++];
        B[i].bn2_b = (const float*)d_in[p++];
        B[i].bn2_g = (const float*)d_in[p++];
        B[i].ln_b  = (const float*)d_in[p++];
        B[i].ln_g  = (const float*)d_in[p++];
        B[i].w1    = (const float*)d_in[p++];
        B[i].w2    = (const float*)d_in[p++];
    }
    const float* fc_b = (const float*)d_in[p++];
    const float* fc_w = (const float*)d_in[p++];

    // ---- workspace carve ----
    char*  w   = (char*)d_ws;
    size_t off = 0;
    auto alloc = [&](size_t bytes) -> void* {
        void* q = w + off;
        off = (off + bytes + 255) & ~(size_t)255;
        return q;
    };
    int*   cntSrc   = (int*)alloc((size_t)N_NODES * 4);
    int*   cntDst   = (int*)alloc((size_t)N_NODES * 4);
    int*   rowstart = (int*)alloc((size_t)(N_NODES + 1) * 4);
    int*   cursor   = (int*)alloc((size_t)N_NODES * 4);
    int*   esrc     = (int*)alloc((size_t)N_EDGES * 4);
    float* ew       = (float*)alloc((size_t)N_EDGES * 4);
    float* dis      = (float*)alloc((size_t)N_NODES * 4);
    float* h        = (float*)alloc((size_t)N_NODES * DH * 4);
    float* t1       = (float*)alloc((size_t)N_NODES * DH * 4);
    float* t2       = (float*)alloc((size_t)N_NODES * DH * 4);
    float* yb       = (float*)alloc((size_t)N_NODES * DH * 4);
    float* zb       = (float*)alloc((size_t)N_NODES * DH * 4);
    float* stats    = (float*)alloc(512 * 4);
    float* bsum = stats, *bsq = stats + 128, *bscale = stats + 256, *bshift = stats + 384;

    const int TB = 256;
    const int gN   = (N_NODES + TB - 1) / TB;
    const int gE   = (N_EDGES + TB - 1) / TB;
    const int gRow = (N_NODES + 7) / 8;            // wave-per-row kernels
    const int gGemm = (N_NODES + 63) / 64;         // 64 rows per block (MTILES=4)
    const int gStat = (N_NODES + 127) / 128;
    const int n4    = N_NODES * DH / 4;
    const int gEl   = (n4 + TB - 1) / TB;

    // ---- CSR build (per call, deterministic) ----
    zero_i32_kernel<<<gN, TB, 0, stream>>>(cntSrc, N_NODES);
    zero_i32_kernel<<<gN, TB, 0, stream>>>(cntDst, N_NODES);
    zero_i32_kernel<<<gN, TB, 0, stream>>>(cursor, N_NODES);
    edge_hist_kernel<<<gE, TB, 0, stream>>>(src, dst, cntSrc, cntDst, N_EDGES);
    dis_kernel<<<gN, TB, 0, stream>>>(cntSrc, dis, N_NODES);
    scan_kernel<<<1, 256, 0, stream>>>(cntDst, rowstart, N_NODES);
    edge_scatter_kernel<<<gE, TB, 0, stream>>>(src, dst, dis, rowstart, cursor, esrc, ew, N_EDGES);

    auto run_block = [&](const float* hin, const BlockParams& bp) {
        // conv1
        spmm_csr_kernel<<<gRow, TB, 0, stream>>>(rowstart, esrc, ew, hin, t1, N_NODES);
        spmm_csr_kernel<<<gRow, TB, 0, stream>>>(rowstart, esrc, ew, t1, t2, N_NODES);
        cheb_gemm_kernel<3, 128, 4><<<gGemm, 256, 0, stream>>>(hin, t1, t2, bp.w1, bp.b1, yb,
                                                               N_NODES);
        // BN1 + relu (in place on yb)
        zero_f32_kernel<<<1, 256, 0, stream>>>(bsum, 256);
        bn_stats_kernel<<<gStat, 256, 0, stream>>>(yb, bsum, bsq, N_NODES);
        bn_finalize_kernel<<<1, 128, 0, stream>>>(bsum, bsq, bp.bn1_g, bp.bn1_b, bscale, bshift,
                                                  1.0f / (float)N_NODES);
        bn_relu_kernel<<<gEl, TB, 0, stream>>>(yb, bscale, bshift, n4);
        // conv2
        spmm_csr_kernel<<<gRow, TB, 0, stream>>>(rowstart, esrc, ew, yb, t1, N_NODES);
        spmm_csr_kernel<<<gRow, TB, 0, stream>>>(rowstart, esrc, ew, t1, t2, N_NODES);
        cheb_gemm_kernel<3, 128, 4><<<gGemm, 256, 0, stream>>>(yb, t1, t2, bp.w2, bp.b2, zb,
                                                               N_NODES);
        // BN2 + residual + LN + relu -> h
        zero_f32_kernel<<<1, 256, 0, stream>>>(bsum, 256);
        bn_stats_kernel<<<gStat, 256, 0, stream>>>(zb, bsum, bsq, N_NODES);
        bn_finalize_kernel<<<1, 128, 0, stream>>>(bsum, bsq, bp.bn2_g, bp.bn2_b, bscale, bshift,
                                                  1.0f / (float)N_NODES);
        bn_res_ln_relu_kernel<<<gRow, TB, 0, stream>>>(zb, hin, bscale, bshift, bp.ln_g, bp.ln_b,
                                                       h, N_NODES);
    };

    run_block(x, B[0]);
    run_block(h, B[1]);
    run_block(h, B[2]);

    // final FC: out = h @ fc_w + fc_b   (K=128, 64 cols)
    cheb_gemm_kernel<1, 64, 4><<<gGemm, 128, 0, stream>>>(h, nullptr, nullptr, fc_w, fc_b,
                                                          (float*)d_out, N_NODES);
}